// PhysicsTopK_85916525789288
// MI455X (gfx1250) — compile-verified
//
#include <hip/hip_runtime.h>
#include <hip/hip_bf16.h>

// Segmented top-k (straight-through) for MI455X / gfx1250.
//
//  Kernel 1: 1025 binary searches over sorted edge_batch -> segment starts.
//  Kernel 2: one workgroup (256 thr = 8 wave32) per graph:
//    - stage segment float bits into LDS once (48KB; counts ~N(8192,90) so
//      12288 capacity is ~45 sigma headroom; global fallback kept),
//    - 4-pass radix select (256-bin LDS histogram per byte plane); the bin
//      search is a PARALLEL 256-wide suffix scan, not a serial loop,
//    - marking: common case is barrier-free (all ties at threshold selected);
//      ordered wave32-ballot rank pass only when ties straddle rank k,
//    - out[e] = ((sel ? 1.f : 0.f) - p) + p   (exact STE float sequence).
//
// Traffic: probs read once (staged to LDS), out written once -> ~64MB total,
// ~2.8us floor at 23.3 TB/s. All selection passes hit LDS, not HBM.

#define NUM_GRAPHS 1024
#define BLK 256
#define NWAVES (BLK / 32)
#define LDS_CAP 12288  // floats staged per segment (48KB)

__global__ void seg_starts_kernel(const int* __restrict__ edge_batch, int E,
                                  int* __restrict__ starts) {
    int g = blockIdx.x * blockDim.x + threadIdx.x;
    if (g > NUM_GRAPHS) return;
    if (g == NUM_GRAPHS) { starts[g] = E; return; }
    // lower_bound: first index with edge_batch[idx] >= g
    int lo = 0, hi = E;
    while (lo < hi) {
        int mid = (lo + hi) >> 1;
        if (edge_batch[mid] < g) lo = mid + 1; else hi = mid;
    }
    starts[g] = lo;
}

__global__ void __launch_bounds__(BLK)
topk_select_kernel(const float* __restrict__ probs,
                   const int* __restrict__ starts,
                   float* __restrict__ out) {
    __shared__ unsigned sBuf[LDS_CAP];
    __shared__ int      sHist[256];
    __shared__ int      sSuf[257];     // inclusive suffix sums; [256] = 0 sentinel
    __shared__ unsigned sPrefix;
    __shared__ unsigned sMask;
    __shared__ int      sRemaining;
    __shared__ int      sTieCount;
    __shared__ int      sWaveCnt[NWAVES];
    __shared__ int      sTieBase;

    const int g = blockIdx.x;
    const int s = starts[g];
    const int c = starts[g + 1] - s;
    if (c == 0) return;  // empty segment

    const int tid = threadIdx.x;

    // k = max(round_half_even(0.3f * c), 1); rintf == nearest-even (default mode)
    int k = (int)rintf(0.3f * (float)c);
    if (k < 1) k = 1;
    const bool selectAll = (k >= c);       // block-uniform
    const bool useLds    = (c <= LDS_CAP); // block-uniform (always true in practice)

    // ---- Stage segment bits into LDS (probs >= 0 => uint order == float order)
    if (useLds) {
        for (int i = tid; i < c; i += BLK) {
            if (i + 4 * BLK < c)
                __builtin_prefetch(&probs[s + i + 4 * BLK], 0, 3); // global_prefetch_b8
            sBuf[i] = __float_as_uint(probs[s + i]);
        }
    }
    __syncthreads();

    if (selectAll) {
        for (int i = tid; i < c; i += BLK) {
            const float p = useLds ? __uint_as_float(sBuf[i]) : probs[s + i];
            out[s + i] = (1.0f - p) + p;
        }
        return;
    }

    // ---- 4-pass radix select for the k-th largest bit pattern
    if (tid == 0) { sPrefix = 0u; sMask = 0u; sRemaining = k; sSuf[256] = 0; }
    __syncthreads();
    for (int pass = 0; pass < 4; ++pass) {
        const int shift = 24 - 8 * pass;
        sHist[tid] = 0;
        __syncthreads();
        const unsigned prefix = sPrefix;
        const unsigned mask   = sMask;
        for (int i = tid; i < c; i += BLK) {
            const unsigned v = useLds ? sBuf[i] : __float_as_uint(probs[s + i]);
            if ((v & mask) == prefix)
                atomicAdd(&sHist[(v >> shift) & 255u], 1);
        }
        // parallel inclusive suffix scan over 256 bins (8 Hillis-Steele steps)
        __syncthreads();
        sSuf[tid] = sHist[tid];
        __syncthreads();
        #pragma unroll
        for (int off = 1; off < 256; off <<= 1) {
            const int v   = sSuf[tid];
            const int add = (tid + off < 256) ? sSuf[tid + off] : 0;
            __syncthreads();
            sSuf[tid] = v + add;
            __syncthreads();
        }
        // the k-th largest lives in the unique bin b with suf[b]>=rem>suf[b+1]
        const int rem     = sRemaining;
        const int sufSelf = sSuf[tid];
        const int sufNext = sSuf[tid + 1];
        __syncthreads();  // all reads of sRemaining/sSuf done before winner writes
        if (sufSelf >= rem && sufNext < rem) {
            sRemaining = rem - (sufSelf - sHist[tid]);  // rank within this bin
            sPrefix    = prefix | ((unsigned)tid << shift);
            sMask      = mask | (255u << shift);
            if (pass == 3) sTieCount = sHist[tid];      // multiplicity of T
        }
        __syncthreads();
    }

    const unsigned T  = sPrefix;     // exact bits of the k-th largest value
    const int      r  = sRemaining;  // #elements == T to select (1..tieCount)
    const int      tc = sTieCount;

    // ---- Common case: every tie at T is selected -> barrier-free marking
    if (r == tc) {
        for (int i = tid; i < c; i += BLK) {
            const unsigned v = useLds ? sBuf[i] : __float_as_uint(probs[s + i]);
            const float p = __uint_as_float(v);
            const float y = (v >= T) ? 1.0f : 0.0f;  // uint cmp == float cmp (non-neg)
            out[s + i] = (y - p) + p;
        }
        return;
    }

    // ---- Rare case: ties straddle rank k -> stable ordered rank pass
    if (tid == 0) sTieBase = 0;
    __syncthreads();

    const int lane = tid & 31;   // wave32
    const int wave = tid >> 5;
    const unsigned laneLtMask = (lane == 0) ? 0u : ((1u << lane) - 1u);
    const int nChunks = (c + BLK - 1) / BLK;

    for (int ch = 0; ch < nChunks; ++ch) {
        const int idx = ch * BLK + tid;
        const bool valid = idx < c;
        unsigned v = 0u;
        if (valid) v = useLds ? sBuf[idx] : __float_as_uint(probs[s + idx]);

        const bool eq = valid && (v == T);
        bool sel      = valid && (v >  T);

        const unsigned m = (unsigned)__ballot(eq);   // low 32 bits on wave32
        if (lane == 0) sWaveCnt[wave] = __popc(m);
        __syncthreads();

        int wpre = 0;
        #pragma unroll
        for (int w = 0; w < NWAVES; ++w)
            if (w < wave) wpre += sWaveCnt[w];
        const int tieRank = sTieBase + wpre + __popc(m & laneLtMask);
        if (eq && tieRank < r) sel = true;

        if (valid) {
            const float p = __uint_as_float(v);
            const float y = sel ? 1.0f : 0.0f;
            out[s + idx] = (y - p) + p;
        }
        __syncthreads();
        if (tid == 0) {
            int tot = 0;
            #pragma unroll
            for (int w = 0; w < NWAVES; ++w) tot += sWaveCnt[w];
            sTieBase += tot;
        }
        __syncthreads();
    }
}

extern "C" void kernel_launch(void* const* d_in, const int* in_sizes, int n_in,
                              void* d_out, int out_size, void* d_ws, size_t ws_size,
                              hipStream_t stream) {
    const float* probs      = (const float*)d_in[0];
    const int*   edge_batch = (const int*)d_in[1];
    const int    E          = in_sizes[0];
    float*       out        = (float*)d_out;
    int*         starts     = (int*)d_ws;   // (NUM_GRAPHS + 1) ints of scratch

    {
        const int n = NUM_GRAPHS + 1;
        seg_starts_kernel<<<(n + BLK - 1) / BLK, BLK, 0, stream>>>(edge_batch, E, starts);
    }
    topk_select_kernel<<<NUM_GRAPHS, BLK, 0, stream>>>(probs, starts, out);
}